// MDNLoss_40011915329853
// MI455X (gfx1250) — compile-verified
//
#include <hip/hip_runtime.h>
#include <hip/hip_bf16.h>
#include <stdint.h>

// ---------------------------------------------------------------------------
// MDN loss (Glow-TTS style) for MI455X / gfx1250.
//   B=32, L=512, T=2048, N_MEL=80.
//   Stage 1: prep    -> Bbig[b][k][l] (k<80: mu*iv, k>=80: -0.5*iv), constL[b][l]
//   Stage 2: gemm    -> P[b][t][l] = sum_k A[t,k]*Bbig[k,l] + constL[l]
//                       via V_WMMA_F32_16X16X4_F32, K=160 split into two
//                       branch-free loops (x-half, x^2-half), 64x16 tile/wave
//   Stage 3: dp scan -> forward algorithm, ONE WAVE per batch, alpha register-
//                       blocked (16 l per lane), shift via __shfl_up, no barriers
//   Stage 4: reduce  -> -mean over batch
// ---------------------------------------------------------------------------

#define NMEL 80
#define BB   32
#define LL   512
#define TT   2048
#define K2   160          // fused K (x-part + x^2-part)
#define NEG_INF -1.0e30f
#define LOG_2PI 1.8378770664093453f

typedef float v2f __attribute__((ext_vector_type(2)));
typedef float v8f __attribute__((ext_vector_type(8)));

// ---------------------------------------------------------------------------
// Stage 1: per-(b,l) preprocessing.
// ---------------------------------------------------------------------------
__global__ void mdn_prep(const float* __restrict__ mu_sigma,
                         float* __restrict__ Bbig,
                         float* __restrict__ constL) {
  int idx = blockIdx.x * blockDim.x + threadIdx.x;   // b*L + l
  if (idx >= BB * LL) return;
  int b = idx / LL;
  int l = idx - b * LL;
  const float* ms = mu_sigma + ((size_t)b * LL + l) * (2 * NMEL);
  float* Bb = Bbig + (size_t)b * K2 * LL;

  float a = 0.0f, sls = 0.0f;
#pragma unroll 4
  for (int m = 0; m < NMEL; ++m) {
    float mu  = ms[m];
    float lsg = ms[NMEL + m];
    float iv  = __expf(-2.0f * lsg);
    a   = fmaf(mu * mu, iv, a);
    sls += lsg;
    Bb[(size_t)m * LL + l]          = mu * iv;
    Bb[(size_t)(NMEL + m) * LL + l] = -0.5f * iv;
  }
  constL[idx] = -0.5f * a - (0.5f * NMEL * LOG_2PI + 0.5f * sls);
}

// ---------------------------------------------------------------------------
// Stage 2: WMMA f32 GEMM producing P[b][t][l] (t-major). One wave computes a
// 64x16 tile (4 sub-tiles along t); the B fragment (k,l-tile) is loaded once
// per k-step and reused by 4 WMMAs. K reduction split into two branch-free
// loops so addressing is affine and no per-iteration predicates are needed.
//
// V_WMMA_F32_16X16X4_F32 operand layouts (ISA 7.12.2, wave32):
//   A 16x4 : lanes 0-15 -> M=0..15, K={0,1}; lanes 16-31 -> M=0..15, K={2,3}.
//   B 4x16 : lanes 0-15 -> N=0..15 rows K={0,1}; lanes 16-31 -> rows K={2,3}.
//   C/D    : VGPR j: lanes 0-15 -> M=j, N=lane; lanes 16-31 -> M=j+8, N=lane-16.
// ---------------------------------------------------------------------------
__global__ __launch_bounds__(256) void mdn_gemm(const float* __restrict__ melspec,
                                                const float* __restrict__ Bbig,
                                                const float* __restrict__ constL,
                                                float* __restrict__ P) {
  const int TTILES = TT / 64;   // 32  (64 rows of t per wave)
  const int LTILES = LL / 16;   // 32

  int wave = (blockIdx.x * blockDim.x + threadIdx.x) >> 5;
  int lane = threadIdx.x & 31;
  int b    = wave / (TTILES * LTILES);
  int rem  = wave - b * (TTILES * LTILES);
  int tt   = rem / LTILES;
  int lt   = rem - tt * LTILES;
  int t0   = tt * 64;
  int l0   = lt * 16;

  int laneM  = lane & 15;   // A row (t offset) / B,D column (l offset)
  int laneHi = lane >> 4;   // selects K pair {0,1} vs {2,3}

  // Per-lane base pointers; both loops advance by 4 rows per k-step.
  const float* melp = melspec + (size_t)b * NMEL * TT
                              + (size_t)(2 * laneHi) * TT + t0 + laneM;
  const float* Bp   = Bbig    + (size_t)b * K2 * LL
                              + (size_t)(2 * laneHi) * LL + l0 + laneM;

  v8f acc0 = {}, acc1 = {}, acc2 = {}, acc3 = {};

  // ---- K half 1: A = x, B = mu*iv (k = 0..79) -----------------------------
#pragma unroll 2
  for (int ks = 0; ks < NMEL; ks += 4) {
    v2f bv, av0, av1, av2, av3;
#pragma unroll
    for (int j = 0; j < 2; ++j) {
      const float* mr = melp + (size_t)(ks + j) * TT;
      bv[j]  = Bp[(size_t)(ks + j) * LL];
      av0[j] = mr[0];
      av1[j] = mr[16];
      av2[j] = mr[32];
      av3[j] = mr[48];
    }
    acc0 = __builtin_amdgcn_wmma_f32_16x16x4_f32(false, av0, false, bv, (short)0, acc0, false, false);
    acc1 = __builtin_amdgcn_wmma_f32_16x16x4_f32(false, av1, false, bv, (short)0, acc1, false, false);
    acc2 = __builtin_amdgcn_wmma_f32_16x16x4_f32(false, av2, false, bv, (short)0, acc2, false, false);
    acc3 = __builtin_amdgcn_wmma_f32_16x16x4_f32(false, av3, false, bv, (short)0, acc3, false, false);
  }

  // ---- K half 2: A = x*x, B = -0.5*iv (k = 80..159) -----------------------
#pragma unroll 2
  for (int ks = 0; ks < NMEL; ks += 4) {
    v2f bv, av0, av1, av2, av3;
#pragma unroll
    for (int j = 0; j < 2; ++j) {
      const float* mr = melp + (size_t)(ks + j) * TT;
      bv[j] = Bp[(size_t)(NMEL + ks + j) * LL];
      float x0 = mr[0], x1 = mr[16], x2 = mr[32], x3 = mr[48];
      av0[j] = x0 * x0;
      av1[j] = x1 * x1;
      av2[j] = x2 * x2;
      av3[j] = x3 * x3;
    }
    acc0 = __builtin_amdgcn_wmma_f32_16x16x4_f32(false, av0, false, bv, (short)0, acc0, false, false);
    acc1 = __builtin_amdgcn_wmma_f32_16x16x4_f32(false, av1, false, bv, (short)0, acc1, false, false);
    acc2 = __builtin_amdgcn_wmma_f32_16x16x4_f32(false, av2, false, bv, (short)0, acc2, false, false);
    acc3 = __builtin_amdgcn_wmma_f32_16x16x4_f32(false, av3, false, bv, (short)0, acc3, false, false);
  }

  float cst = constL[(size_t)b * LL + l0 + laneM];         // depends only on column
  float* Pb = P + ((size_t)b * TT + t0) * LL + l0;
#pragma unroll
  for (int j = 0; j < 8; ++j) {
    int row = j + 8 * laneHi;                              // M index within sub-tile
    Pb[(size_t)(row     ) * LL + laneM] = acc0[j] + cst;
    Pb[(size_t)(row + 16) * LL + laneM] = acc1[j] + cst;
    Pb[(size_t)(row + 32) * LL + laneM] = acc2[j] + cst;
    Pb[(size_t)(row + 48) * LL + laneM] = acc3[j] + cst;
  }
}

// ---------------------------------------------------------------------------
// Stage 3: monotonic forward scan — ONE wave32 per batch, zero barriers.
// Lane owns l = lane*16 .. lane*16+15 in registers a[0..15]. Per step:
//   - one __shfl_up carries a[15] across the lane boundary
//   - 15 shifts are pure register moves
//   - column load = 4x float4 (global_load_b128), contiguous, L2-resident
//   - prefetch next column (global_prefetch_b8) to hide L2 latency
// new[l] = logaddexp(a[l], a[l-1]) + P[b][t][l]
// ---------------------------------------------------------------------------
__global__ __launch_bounds__(32) void mdn_scan(const float* __restrict__ P,
                                               const long long* __restrict__ text_len,
                                               const long long* __restrict__ mel_len,
                                               float* __restrict__ slots) {
  int b    = blockIdx.x;
  int lane = threadIdx.x;          // 0..31

  const float* Pb = P + (size_t)b * TT * LL;
  int tl = (int)text_len[b] - 1;   // text index to gather
  int ml = (int)mel_len[b]  - 1;   // time step to gather at
  int tlLane = tl >> 4;            // lane owning l = tl
  int tlSub  = tl & 15;

  float a[16];
#pragma unroll
  for (int j = 0; j < 16; ++j) a[j] = NEG_INF;
  if (lane == 0) a[0] = Pb[0];     // alpha_0[l] = logp[0,0] at l==0 only
  if (ml == 0 && lane == tlLane) { // degenerate gather at t=0
#pragma unroll
    for (int j = 0; j < 16; ++j) if (j == tlSub) slots[b] = a[j];
  }

  for (int t = 1; t < TT; ++t) {
    const float4* col = (const float4*)(Pb + (size_t)t * LL) + lane * 4;
    if (t + 1 < TT)
      __builtin_prefetch(Pb + (size_t)(t + 1) * LL + lane * 16, 0, 0);

    // carry a[15] of lane-1 into this lane's left[0]
    float carry = __shfl_up(a[15], 1, 32);
    if (lane == 0) carry = NEG_INF;

    float4 c0 = col[0], c1 = col[1], c2 = col[2], c3 = col[3];
    float lp[16] = { c0.x, c0.y, c0.z, c0.w, c1.x, c1.y, c1.z, c1.w,
                     c2.x, c2.y, c2.z, c2.w, c3.x, c3.y, c3.z, c3.w };

    float na[16];
#pragma unroll
    for (int j = 0; j < 16; ++j) {
      float left = (j == 0) ? carry : a[j - 1];
      // logaddexp: max + log1p(exp(-|d|)); exact for the -1e30 sentinel too.
      float mx = fmaxf(a[j], left);
      na[j] = mx + log1pf(__expf(-fabsf(a[j] - left))) + lp[j];
    }
#pragma unroll
    for (int j = 0; j < 16; ++j) a[j] = na[j];

    if (t == ml && lane == tlLane) {
#pragma unroll
      for (int j = 0; j < 16; ++j) if (j == tlSub) slots[b] = a[j];
    }
  }
}

// ---------------------------------------------------------------------------
// Stage 4: -mean over batch. Single wave32, shuffle reduction.
// ---------------------------------------------------------------------------
__global__ void mdn_reduce(const float* __restrict__ slots, float* __restrict__ out) {
  float v = slots[threadIdx.x & 31];
#pragma unroll
  for (int off = 16; off > 0; off >>= 1)
    v += __shfl_down(v, off, 32);
  if (threadIdx.x == 0) out[0] = -v / (float)BB;
}

// ---------------------------------------------------------------------------
extern "C" void kernel_launch(void* const* d_in, const int* in_sizes, int n_in,
                              void* d_out, int out_size, void* d_ws, size_t ws_size,
                              hipStream_t stream) {
  (void)in_sizes; (void)n_in; (void)out_size; (void)ws_size;

  const float*     mu_sigma = (const float*)d_in[0];
  const float*     melspec  = (const float*)d_in[1];
  const long long* text_len = (const long long*)d_in[2];
  const long long* mel_len  = (const long long*)d_in[3];
  float*           out      = (float*)d_out;

  // Workspace layout
  float* P      = (float*)d_ws;                         // [B][T][L]  128 MiB (L2-resident)
  float* Bbig   = P      + (size_t)BB * TT * LL;        // [B][160][L] 10 MiB
  float* constL = Bbig   + (size_t)BB * K2 * LL;        // [B][L]
  float* slots  = constL + (size_t)BB * LL;             // [B]

  // Stage 1: prep (B*L threads)
  {
    int n = BB * LL;
    mdn_prep<<<(n + 255) / 256, 256, 0, stream>>>(mu_sigma, Bbig, constL);
  }
  // Stage 2: WMMA GEMM — one wave per 64x16 tile, 8 waves per block
  {
    int waves  = BB * (TT / 64) * (LL / 16);            // 32768
    int blocks = waves / 8;                             // 256 threads/block
    mdn_gemm<<<blocks, 256, 0, stream>>>(melspec, Bbig, constL, P);
  }
  // Stage 3: scan — one wave per batch, no barriers
  mdn_scan<<<BB, 32, 0, stream>>>(P, text_len, mel_len, slots);
  // Stage 4: reduce
  mdn_reduce<<<1, 32, 0, stream>>>(slots, out);
}